// Block_16234976378917
// MI455X (gfx1250) — compile-verified
//
#include <hip/hip_runtime.h>

// ---------------------------------------------------------------------------
// Geometric transformer block for MI455X (gfx1250, wave32, WMMA).
// Dense contractions: V_WMMA_F32_16X16X4_F32 (fp32 matrix pipe).
// Tile staging: GLOBAL_LOAD_ASYNC_TO_LDS_{B32,B128} + s_wait_asynccnt (CDNA5).
// ---------------------------------------------------------------------------

typedef float v2f __attribute__((ext_vector_type(2)));
typedef float v8f __attribute__((ext_vector_type(8)));

constexpr int A_ = 128, Vd_ = 64, RANK_ = 32, KQ_ = 64, H_ = 4, KC_ = 7;
constexpr int B_ = 8, N_ = 1024;
constexpr long BN_ = (long)B_ * N_;

#define DEV static __device__ __forceinline__

DEV float leaky_(float x) { return x > 0.f ? x : 0.2f * x; }

// wave-local LDS fence (orders plain ds_store -> wmma ds_load within a wave)
DEV void lds_fence() { asm volatile("s_wait_dscnt 0" ::: "memory"); }

// Async global->LDS DMA (CDNA5). LDS operand is the per-lane LDS byte offset
// (low 32 bits of the flat shared address); address operand is 64-bit VGPR pair.
DEV void async_b128(void* lds, const void* g) {
  asm volatile("global_load_async_to_lds_b128 %0, %1, off"
               :: "v"((unsigned)(unsigned long)lds), "v"(g) : "memory");
}
DEV void async_b32(void* lds, const void* g) {
  asm volatile("global_load_async_to_lds_b32 %0, %1, off"
               :: "v"((unsigned)(unsigned long)lds), "v"(g) : "memory");
}
DEV void wait_async() { asm volatile("s_wait_asynccnt 0" ::: "memory"); }

// D(16x16) += A(16xK) * B(KxN=16), A row-major (ldA), B given as B^T row-major (ldB).
// f32 WMMA operand layout: lane holds A[m = lane&15, k0 + 2*(lane>>4) + {0,1}].
DEV v8f wmma16(const float* lA, int ldA, const float* lBt, int ldB, int Klen, v8f acc) {
  const int lane = threadIdx.x & 31;
  const int m = lane & 15;
  const int kk = (lane >> 4) << 1;
  for (int k0 = 0; k0 < Klen; k0 += 4) {
    v2f a = { lA[m * ldA + k0 + kk], lA[m * ldA + k0 + kk + 1] };
    v2f b = { lBt[m * ldB + k0 + kk], lBt[m * ldB + k0 + kk + 1] };
    acc = __builtin_amdgcn_wmma_f32_16x16x4_f32(false, a, false, b, (short)0, acc, false, false);
  }
  return acc;
}

// Same but B in natural row-major (K x ncols) layout (column gather).
DEV v8f wmma16_rowB(const float* lA, int ldA, const float* lB, int ldB, int col0, int Klen, v8f acc) {
  const int lane = threadIdx.x & 31;
  const int m = lane & 15;
  const int kk = (lane >> 4) << 1;
  for (int k0 = 0; k0 < Klen; k0 += 4) {
    v2f a = { lA[m * ldA + k0 + kk], lA[m * ldA + k0 + kk + 1] };
    v2f b = { lB[(k0 + kk) * ldB + col0 + m], lB[(k0 + kk + 1) * ldB + col0 + m] };
    acc = __builtin_amdgcn_wmma_f32_16x16x4_f32(false, a, false, b, (short)0, acc, false, false);
  }
  return acc;
}

// ------------------------- edge features (VALU, tiny) ----------------------
__global__ void edge_feat_kernel(const float* __restrict__ pos0, const float* __restrict__ pos1,
                                 const float* We1, const float* be1,
                                 const float* We2, const float* be2,
                                 const float* Wev, float* ea, float* ev) {
  const int bn = blockIdx.x;
  const int b = bn / (N_ - 1), n = bn % (N_ - 1);
  const float* p0 = pos0 + ((long)b * N_ + n) * 3;
  const float* p1 = pos1 + ((long)b * N_ + n) * 3;
  float d0[3], d1[3];
#pragma unroll
  for (int i = 0; i < 3; i++) { d0[i] = p0[3 + i] - p0[i]; d1[i] = p1[3 + i] - p1[i]; }
  float ef0 = sqrtf(d0[0]*d0[0] + d0[1]*d0[1] + d0[2]*d0[2] + 1e-8f);
  float ef1 = sqrtf(d1[0]*d1[0] + d1[1]*d1[1] + d1[2]*d1[2] + 1e-8f);
  float ef2 = d0[0]*d1[0] + d0[1]*d1[1] + d0[2]*d1[2];
  __shared__ float hsm[A_];
  const int o = threadIdx.x;
  hsm[o] = leaky_(ef0 * We1[o] + ef1 * We1[A_ + o] + ef2 * We1[2 * A_ + o] + be1[o]);
  __syncthreads();
  float acc = be2[o];
  for (int c = 0; c < A_; c++) acc += hsm[c] * We2[c * A_ + o];
  ea[((long)b * N_ + n) * A_ + o] = acc;
  if (o < Vd_) {
#pragma unroll
    for (int i = 0; i < 3; i++)
      ev[(((long)b * N_ + n) * Vd_ + o) * 3 + i] = d0[i] * Wev[o] + d1[i] * Wev[Vd_ + o];
  }
}

__global__ void emb_kernel(const float* __restrict__ pos0, const float* __restrict__ pos1,
                           const float* __restrict__ ea, const float* __restrict__ ev,
                           const float* Wn1, const float* bn1, const float* Wn2, const float* bn2,
                           const float* Wnv, float* emba, float* embv) {
  const int bn = blockIdx.x;
  const int b = bn / N_, n = bn % N_;
  const int o = threadIdx.x;
  float dv[3];
#pragma unroll
  for (int i = 0; i < 3; i++) dv[i] = pos1[(long)bn * 3 + i] - pos0[(long)bn * 3 + i];
  float nf = sqrtf(dv[0]*dv[0] + dv[1]*dv[1] + dv[2]*dv[2] + 1e-8f);
  __shared__ float hsm[A_];
  hsm[o] = leaky_(nf * Wn1[o] + bn1[o]);
  __syncthreads();
  float acc = bn2[o];
  for (int c = 0; c < A_; c++) acc += hsm[c] * Wn2[c * A_ + o];
  float l = (n >= 1) ? ea[((long)b * N_ + (n - 1)) * A_ + o] : 0.f;
  float r = (n <= N_ - 2) ? ea[((long)b * N_ + n) * A_ + o] : 0.f;
  emba[(long)bn * A_ + o] = 0.5f * (l + r) + acc;
  if (o < Vd_) {
#pragma unroll
    for (int i = 0; i < 3; i++) {
      float lv = (n >= 1) ? ev[(((long)b * N_ + (n - 1)) * Vd_ + o) * 3 + i] : 0.f;
      float rv = (n <= N_ - 2) ? ev[(((long)b * N_ + n) * Vd_ + o) * 3 + i] : 0.f;
      embv[((long)bn * Vd_ + o) * 3 + i] = 0.5f * (lv + rv) + dv[i] * Wnv[o];
    }
  }
}

// ------------------------- conv1d (K=7) via WMMA ----------------------------
__global__ void __launch_bounds__(128)
conv_a_kernel(const float* __restrict__ xin, const float* __restrict__ Kw,
              const float* __restrict__ bias, const float* __restrict__ addb,
              float* __restrict__ out) {
  const int b = blockIdx.x / (N_ / 16);
  const int n0 = (blockIdx.x % (N_ / 16)) * 16;
  const int w = threadIdx.x >> 5, lane = threadIdx.x & 31;
  const int col0 = (blockIdx.y * 4 + w) * 16;
  __shared__ float xs[22][A_];       // rows n0-3..n0+18, zero padded
  __shared__ float bt[4][16][A_];    // per-wave B^T tile
  // async DMA the input window (b128 = 4 floats/lane/instr); zero-pad edges via DS
  for (int idx4 = threadIdx.x; idx4 < 22 * (A_ / 4); idx4 += 128) {
    int rr = idx4 >> 5, i4 = (idx4 & 31) * 4;
    int nn = n0 - 3 + rr;
    if (nn >= 0 && nn < N_)
      async_b128(&xs[rr][i4], &xin[((long)b * N_ + nn) * A_ + i4]);
    else { xs[rr][i4] = 0.f; xs[rr][i4 + 1] = 0.f; xs[rr][i4 + 2] = 0.f; xs[rr][i4 + 3] = 0.f; }
  }
  wait_async();
  __syncthreads();
  v8f acc = {0, 0, 0, 0, 0, 0, 0, 0};
  for (int t = 0; t < KC_; t++) {
    // gathered weight tile: per-element async DMA
    for (int idx = lane; idx < 16 * A_; idx += 32) {
      int nn = idx >> 7, i = idx & 127;
      async_b32(&bt[w][nn][i], &Kw[((long)t * A_ + i) * A_ + col0 + nn]);
    }
    wait_async();
    acc = wmma16(&xs[t][0], A_, &bt[w][0][0], A_, A_, acc);
  }
  const int half = lane >> 4, nloc = lane & 15;
  const int col = col0 + nloc;
  const float bv = bias ? bias[col] : 0.f;
#pragma unroll
  for (int r = 0; r < 8; r++) {
    long idx = ((long)b * N_ + n0 + r + 8 * half) * A_ + col;
    float v = acc[r] + bv;
    if (addb) v += addb[idx];
    out[idx] = v;
  }
}

__global__ void __launch_bounds__(128)
conv_v_kernel(const float* __restrict__ xv, const float* __restrict__ Kw,
              const float* __restrict__ addb, float* __restrict__ out) {
  const int b = blockIdx.x / (N_ / 16);
  const int n0 = (blockIdx.x % (N_ / 16)) * 16;
  const int w = threadIdx.x >> 5, lane = threadIdx.x & 31;
  __shared__ float xs[3][22][Vd_];
  __shared__ float bt[4][16][Vd_];
  for (int idx = threadIdx.x; idx < 3 * 22 * Vd_; idx += 128) {
    int comp = idx / (22 * Vd_);
    int rem = idx % (22 * Vd_);
    int rr = rem / Vd_, c = rem % Vd_;
    int nn = n0 - 3 + rr;
    if (nn >= 0 && nn < N_)
      async_b32(&xs[comp][rr][c], &xv[(((long)b * N_ + nn) * Vd_ + c) * 3 + comp]);
    else
      xs[comp][rr][c] = 0.f;
  }
  wait_async();
  __syncthreads();
  for (int tile = w; tile < 12; tile += 4) {
    const int comp = tile >> 2, col0 = (tile & 3) * 16;
    v8f acc = {0, 0, 0, 0, 0, 0, 0, 0};
    for (int t = 0; t < KC_; t++) {
      for (int idx = lane; idx < 16 * Vd_; idx += 32) {
        int nn = idx >> 6, c = idx & 63;
        async_b32(&bt[w][nn][c], &Kw[((long)t * Vd_ + c) * Vd_ + col0 + nn]);
      }
      wait_async();
      acc = wmma16(&xs[comp][t][0], Vd_, &bt[w][0][0], Vd_, Vd_, acc);
    }
    const int half = lane >> 4, nloc = lane & 15;
#pragma unroll
    for (int r = 0; r < 8; r++) {
      long idx = (((long)b * N_ + n0 + r + 8 * half) * Vd_ + col0 + nloc) * 3 + comp;
      float v = acc[r];
      if (addb) v += addb[idx];
      out[idx] = v;
    }
  }
}

// ------------------------- generic dense GEMM via WMMA ----------------------
__global__ void __launch_bounds__(128)
gemm_kernel(const float* __restrict__ in, int ldin,
            const float* __restrict__ W, const float* __restrict__ bias,
            float* __restrict__ out, int ldout,
            int Kdim, int ncols, int act, int accum) {
  const long row0 = (long)blockIdx.x * 16;
  const int w = threadIdx.x >> 5, lane = threadIdx.x & 31;
  __shared__ float as[16][A_];
  __shared__ float bt[4][16][A_];
  const int k4n = Kdim >> 2;
  for (int idx4 = threadIdx.x; idx4 < 16 * k4n; idx4 += 128) {
    int r = idx4 / k4n, k4 = (idx4 % k4n) * 4;
    async_b128(&as[r][k4], &in[(row0 + r) * ldin + k4]);
  }
  wait_async();
  __syncthreads();
  const int ntiles = ncols >> 4;
  for (int tile = w; tile < ntiles; tile += 4) {
    const int col0 = tile * 16;
    for (int idx = lane; idx < 16 * Kdim; idx += 32) {
      int nn = idx / Kdim, kk = idx % Kdim;
      async_b32(&bt[w][nn][kk], &W[(long)kk * ncols + col0 + nn]);
    }
    wait_async();
    v8f acc = {0, 0, 0, 0, 0, 0, 0, 0};
    acc = wmma16(&as[0][0], A_, &bt[w][0][0], A_, Kdim, acc);
    const int half = lane >> 4, nloc = lane & 15;
    const int col = col0 + nloc;
    const float bv = bias ? bias[col] : 0.f;
#pragma unroll
    for (int r = 0; r < 8; r++) {
      long idx = (row0 + r + 8 * half) * ldout + col;
      float v = acc[r] + bv;
      if (act) v = leaky_(v);
      if (accum) out[idx] += v; else out[idx] = v;
    }
  }
}

// ------------------------- vector-channel linear (VALU) ---------------------
__global__ void vlin_kernel(const float* __restrict__ vin, const float* __restrict__ W,
                            float* __restrict__ out, int cin, int cout, int dovroot, int accum) {
  const long bn = blockIdx.x;
  const int t = threadIdx.x;  // 64
  __shared__ float vsm[64][3];
  if (t < cin) {
    float v0 = vin[(bn * cin + t) * 3 + 0];
    float v1 = vin[(bn * cin + t) * 3 + 1];
    float v2 = vin[(bn * cin + t) * 3 + 2];
    if (dovroot) {
      float nn = sqrtf(v0 * v0 + v1 * v1 + v2 * v2 + 1e-8f);
      float s = rsqrtf(1.f + nn);
      v0 *= s; v1 *= s; v2 *= s;
    }
    vsm[t][0] = v0; vsm[t][1] = v1; vsm[t][2] = v2;
  }
  __syncthreads();
  if (t < cout) {
    float a0 = 0, a1 = 0, a2 = 0;
    for (int c = 0; c < cin; c++) {
      float wc = W[c * cout + t];
      a0 += vsm[c][0] * wc; a1 += vsm[c][1] * wc; a2 += vsm[c][2] * wc;
    }
    long idx = (bn * cout + t) * 3;
    if (accum) { out[idx] += a0; out[idx + 1] += a1; out[idx + 2] += a2; }
    else       { out[idx]  = a0; out[idx + 1]  = a1; out[idx + 2]  = a2; }
  }
}

__global__ void rank_mix_kernel(const float* __restrict__ ar, const float* __restrict__ v1,
                                const float* __restrict__ v2, float* pa_in, float* pv_in) {
  const long bn = blockIdx.x;
  const int c = threadIdx.x;  // 32
  long vi = (bn * RANK_ + c) * 3;
  float x0 = v1[vi], x1 = v1[vi + 1], x2 = v1[vi + 2];
  float y0 = v2[vi], y1 = v2[vi + 1], y2 = v2[vi + 2];
  pa_in[bn * RANK_ + c] = x0 * y0 + x1 * y1 + x2 * y2;
  float a = ar[bn * RANK_ + c];
  pv_in[vi] = a * x0; pv_in[vi + 1] = a * x1; pv_in[vi + 2] = a * x2;
}

// ------------------------- group norms --------------------------------------
__global__ void sgn_kernel(const float* __restrict__ in1, const float* __restrict__ in2,
                           const float* g, const float* bb,
                           const float* __restrict__ base, float* __restrict__ dst) {
  const long bn = blockIdx.x;
  const int o = threadIdx.x;  // 128
  __shared__ float xs[A_];
  long idx = bn * A_ + o;
  float x = in1[idx] + (in2 ? in2[idx] : 0.f);
  xs[o] = x;
  __syncthreads();
  const int g0 = o & ~15;
  float m = 0.f, s = 0.f;
  for (int j = 0; j < 16; j++) { float v = xs[g0 + j]; m += v; s += v * v; }
  m *= 0.0625f; s = s * 0.0625f - m * m;
  float val = (x - m) * rsqrtf(s + 1e-5f) * g[o] + bb[o];
  float bv = base ? base[idx] : dst[idx];
  dst[idx] = bv + val;
}

__global__ void vgn_kernel(const float* __restrict__ in1, const float* __restrict__ in2,
                           const float* g, const float* __restrict__ base, float* __restrict__ dst) {
  const long bn = blockIdx.x;
  const int c = threadIdx.x;  // 64
  __shared__ float n2s[Vd_];
  long idx = (bn * Vd_ + c) * 3;
  float t0 = in1[idx], t1 = in1[idx + 1], t2 = in1[idx + 2];
  if (in2) { t0 += in2[idx]; t1 += in2[idx + 1]; t2 += in2[idx + 2]; }
  n2s[c] = t0 * t0 + t1 * t1 + t2 * t2;
  __syncthreads();
  const int g0 = c & ~7;
  float m = 0.f;
  for (int j = 0; j < 8; j++) m += n2s[g0 + j];
  float gc = g[c] * rsqrtf(m * 0.125f + 1e-5f);
  float o0 = t0 * gc, o1 = t1 * gc, o2 = t2 * gc;
  if (base) { dst[idx] = base[idx] + o0; dst[idx + 1] = base[idx + 1] + o1; dst[idx + 2] = base[idx + 2] + o2; }
  else      { dst[idx] += o0; dst[idx + 1] += o1; dst[idx + 2] += o2; }
}

// ------------------------- probe points -------------------------------------
__global__ void probes_kernel(const float* __restrict__ ya, const float* __restrict__ yv,
                              const float* __restrict__ pos0, const float* __restrict__ pos1,
                              const float* Wpa_, const float* Wpv_,
                              float* pq, float* pk, float* sqq, float* sqk) {
  const long bn = blockIdx.x;
  const int t = threadIdx.x;  // 32
  __shared__ float lg[8], off[12], pr[2][2][3];
  if (t < 8) {
    float acc = 0;
    for (int c = 0; c < A_; c++) acc += ya[bn * A_ + c] * Wpa_[c * 8 + t];
    lg[t] = acc;
  }
  if (t < 12) {
    int sp = t / 3, i = t % 3;
    float acc = 0;
    for (int c = 0; c < Vd_; c++) acc += yv[(bn * Vd_ + c) * 3 + i] * Wpv_[c * 4 + sp];
    off[t] = acc;
  }
  __syncthreads();
  if (t < 4) {
    int s = t >> 1, p = t & 1;
    float l0 = lg[s * 4 + p * 2 + 0], l1 = lg[s * 4 + p * 2 + 1];
    float mx = fmaxf(l0, l1);
    float e0 = __expf(l0 - mx), e1 = __expf(l1 - mx);
    float inv = 1.f / (e0 + e1);
    for (int i = 0; i < 3; i++) {
      float base = e0 * inv * pos0[bn * 3 + i] + e1 * inv * pos1[bn * 3 + i];
      pr[s][p][i] = base + off[(s * 2 + p) * 3 + i];
    }
  }
  __syncthreads();
  if (t == 0) {
    float aq = 0, ak = 0;
    for (int p = 0; p < 2; p++)
      for (int i = 0; i < 3; i++) {
        float vk = pr[0][p][i], vq = pr[1][p][i];
        pk[(bn * 2 + p) * 3 + i] = vk; pq[(bn * 2 + p) * 3 + i] = vq;
        ak += vk * vk; aq += vq * vq;
      }
    sqk[bn] = ak; sqq[bn] = aq;
  }
}

// ------------------------- flash attention with geometric bias --------------
__global__ void __launch_bounds__(128)
attn_kernel(const float* __restrict__ q, const float* __restrict__ k,
            const float* __restrict__ va, const float* __restrict__ vv,
            const float* __restrict__ pq, const float* __restrict__ pk,
            const float* __restrict__ sqq, const float* __restrict__ sqk,
            float* __restrict__ oa_pre, float* __restrict__ ov_pre) {
  const int qt = blockIdx.x & 63;
  const int h = (blockIdx.x >> 6) & 3;
  const int b = blockIdx.x >> 8;
  const int i0 = qt * 16;
  const int w = threadIdx.x >> 5, lane = threadIdx.x & 31;
  const int half = lane >> 4, nloc = lane & 15;
  const float invr = 1.f / (float)(1 << (2 * h));  // R0SQ = {1,4,16,64}

  __shared__ float qs[16][KQ_];
  __shared__ float pqs[16][6];
  __shared__ float sqq_s[16];
  __shared__ float ks[4][16][KQ_];
  __shared__ float pks[4][16][6];
  __shared__ float sqk_s[4][16];
  __shared__ float ps[4][16][16];
  __shared__ float vsm[4][16][80];   // cols: 0..31 = va(h), 32..79 = vv(h) d*3+c
  __shared__ float red_m[4][16], red_l[4][16];
  __shared__ float osh[16][80];

  for (int idx4 = threadIdx.x; idx4 < 16 * (KQ_ / 4); idx4 += 128) {
    int r = idx4 >> 4, c4 = (idx4 & 15) * 4;
    async_b128(&qs[r][c4], &q[((long)b * N_ + i0 + r) * (H_ * KQ_) + h * KQ_ + c4]);
  }
  for (int idx = threadIdx.x; idx < 16 * 6; idx += 128) {
    int r = idx / 6, e = idx % 6;
    pqs[r][e] = pq[((long)b * N_ + i0 + r) * 6 + e];
  }
  if (threadIdx.x < 16) sqq_s[threadIdx.x] = sqq[(long)b * N_ + i0 + threadIdx.x];
  for (int idx = threadIdx.x; idx < 16 * 80; idx += 128) osh[idx / 80][idx % 80] = 0.f;
  wait_async();
  __syncthreads();

  float mrun[8], lrun[8];
  v8f oacc[5];
#pragma unroll
  for (int r = 0; r < 8; r++) { mrun[r] = -1e30f; lrun[r] = 0.f; }
#pragma unroll
  for (int ct = 0; ct < 5; ct++) { v8f z = {0, 0, 0, 0, 0, 0, 0, 0}; oacc[ct] = z; }

  for (int jt = w; jt < N_ / 16; jt += 4) {
    const int j0 = jt * 16;
    for (int idx4 = lane; idx4 < 16 * (KQ_ / 4); idx4 += 32) {
      int r = idx4 >> 4, c4 = (idx4 & 15) * 4;
      async_b128(&ks[w][r][c4], &k[((long)b * N_ + j0 + r) * (H_ * KQ_) + h * KQ_ + c4]);
    }
    for (int idx = lane; idx < 16 * 6; idx += 32) {
      int r = idx / 6, e = idx % 6;
      pks[w][r][e] = pk[((long)b * N_ + j0 + r) * 6 + e];
    }
    if (lane < 16) sqk_s[w][lane] = sqk[(long)b * N_ + j0 + lane];
    wait_async();
    lds_fence();

    v8f s = {0, 0, 0, 0, 0, 0, 0, 0};
    s = wmma16(&qs[0][0], KQ_, &ks[w][0][0], KQ_, KQ_, s);

    float scale[8];
#pragma unroll
    for (int r = 0; r < 8; r++) {
      const int rl = r + 8 * half;
      float dot = 0.f;
#pragma unroll
      for (int e = 0; e < 6; e++) dot += pqs[rl][e] * pks[w][nloc][e];
      float d2 = sqq_s[rl] + sqk_s[w][nloc] - 2.f * dot;
      float logit = s[r] * 0.125f - d2 * invr;
      float rm = logit;
      rm = fmaxf(rm, __shfl_xor(rm, 1, 32));
      rm = fmaxf(rm, __shfl_xor(rm, 2, 32));
      rm = fmaxf(rm, __shfl_xor(rm, 4, 32));
      rm = fmaxf(rm, __shfl_xor(rm, 8, 32));
      float mnew = fmaxf(mrun[r], rm);
      scale[r] = __expf(mrun[r] - mnew);
      float pe = __expf(logit - mnew);
      float rs = pe;
      rs += __shfl_xor(rs, 1, 32);
      rs += __shfl_xor(rs, 2, 32);
      rs += __shfl_xor(rs, 4, 32);
      rs += __shfl_xor(rs, 8, 32);
      lrun[r] = lrun[r] * scale[r] + rs;
      mrun[r] = mnew;
      ps[w][rl][nloc] = pe;
    }
#pragma unroll
    for (int ct = 0; ct < 5; ct++)
#pragma unroll
      for (int r = 0; r < 8; r++) oacc[ct][r] *= scale[r];

    // V tile: va rows are contiguous (b128); vv part is a strided gather (b32)
    for (int idx4 = lane; idx4 < 16 * 8; idx4 += 32) {
      int r = idx4 >> 3, c4 = (idx4 & 7) * 4;
      async_b128(&vsm[w][r][c4], &va[((long)b * N_ + j0 + r) * A_ + h * 32 + c4]);
    }
    for (int idx = lane; idx < 16 * 48; idx += 32) {
      int r = idx / 48, dc = idx % 48;
      int d = dc / 3, cc = dc % 3;
      async_b32(&vsm[w][r][32 + dc],
                &vv[(((long)b * N_ + j0 + r) * Vd_ + h * 16 + d) * 3 + cc]);
    }
    wait_async();
    lds_fence();
#pragma unroll
    for (int ct = 0; ct < 5; ct++)
      oacc[ct] = wmma16_rowB(&ps[w][0][0], 16, &vsm[w][0][0], 80, ct * 16, 16, oacc[ct]);
  }

  // combine the 4 split-softmax partials
#pragma unroll
  for (int r = 0; r < 8; r++) { int rl = r + 8 * half; red_m[w][rl] = mrun[r]; red_l[w][rl] = lrun[r]; }
  __syncthreads();
  float gm[8];
#pragma unroll
  for (int r = 0; r < 8; r++) {
    int rl = r + 8 * half;
    float m = fmaxf(fmaxf(red_m[0][rl], red_m[1][rl]), fmaxf(red_m[2][rl], red_m[3][rl]));
    gm[r] = m;
  }
  for (int w2 = 0; w2 < 4; w2++) {
    if (w == w2) {
#pragma unroll
      for (int ct = 0; ct < 5; ct++)
#pragma unroll
        for (int r = 0; r < 8; r++) {
          int rl = r + 8 * half;
          osh[rl][ct * 16 + nloc] += __expf(mrun[r] - gm[r]) * oacc[ct][r];
        }
    }
    __syncthreads();
  }
  for (int idx = threadIdx.x; idx < 16 * 80; idx += 128) {
    int r = idx / 80, c = idx % 80;
    float M = fmaxf(fmaxf(red_m[0][r], red_m[1][r]), fmaxf(red_m[2][r], red_m[3][r]));
    float L = red_l[0][r] * __expf(red_m[0][r] - M) + red_l[1][r] * __expf(red_m[1][r] - M) +
              red_l[2][r] * __expf(red_m[2][r] - M) + red_l[3][r] * __expf(red_m[3][r] - M);
    float val = osh[r][c] / L;
    long n = (long)b * N_ + i0 + r;
    if (c < 32) oa_pre[n * A_ + h * 32 + c] = val;
    else { int dc = c - 32; int d = dc / 3, cc = dc % 3;
           ov_pre[(n * Vd_ + h * 16 + d) * 3 + cc] = val; }
  }
}

// ---------------------------------------------------------------------------
extern "C" void kernel_launch(void* const* d_in, const int* in_sizes, int n_in,
                              void* d_out, int out_size, void* d_ws, size_t ws_size,
                              hipStream_t stream) {
  (void)in_sizes; (void)n_in; (void)out_size; (void)ws_size;
  const float* pos0 = (const float*)d_in[0];
  const float* pos1 = (const float*)d_in[1];
  const float* x_a  = (const float*)d_in[2];
  const float* x_v  = (const float*)d_in[3];
  int pi = 4;
  const float* We1 = (const float*)d_in[pi++]; const float* be1 = (const float*)d_in[pi++];
  const float* We2 = (const float*)d_in[pi++]; const float* be2 = (const float*)d_in[pi++];
  const float* Wev = (const float*)d_in[pi++];
  const float* Wn1 = (const float*)d_in[pi++]; const float* bn1 = (const float*)d_in[pi++];
  const float* Wn2 = (const float*)d_in[pi++]; const float* bn2 = (const float*)d_in[pi++];
  const float* Wnv = (const float*)d_in[pi++];
  const float* K0a = (const float*)d_in[pi++]; const float* b0a = (const float*)d_in[pi++];
  const float* K0v = (const float*)d_in[pi++];
  const float* Wl1 = (const float*)d_in[pi++]; const float* bl1 = (const float*)d_in[pi++];
  const float* Wl2 = (const float*)d_in[pi++]; const float* bl2 = (const float*)d_in[pi++];
  const float* Wl3 = (const float*)d_in[pi++]; const float* bl3 = (const float*)d_in[pi++];
  const float* Wv1 = (const float*)d_in[pi++]; const float* Wv2 = (const float*)d_in[pi++];
  const float* Wv3 = (const float*)d_in[pi++];
  const float* War = (const float*)d_in[pi++]; const float* bar = (const float*)d_in[pi++];
  const float* Wvr1 = (const float*)d_in[pi++]; const float* Wvr2 = (const float*)d_in[pi++];
  const float* Wpa = (const float*)d_in[pi++]; const float* bpa = (const float*)d_in[pi++];
  const float* Wpv = (const float*)d_in[pi++];
  const float* glra = (const float*)d_in[pi++]; const float* blra = (const float*)d_in[pi++];
  const float* glrv = (const float*)d_in[pi++];
  const float* K1a = (const float*)d_in[pi++]; const float* b1a = (const float*)d_in[pi++];
  const float* K1v = (const float*)d_in[pi++];
  const float* Wprobe_a = (const float*)d_in[pi++]; const float* Wprobe_v = (const float*)d_in[pi++];
  const float* Wq = (const float*)d_in[pi++]; const float* Wk = (const float*)d_in[pi++];
  const float* Wva = (const float*)d_in[pi++]; const float* Wvv = (const float*)d_in[pi++];
  const float* Woa = (const float*)d_in[pi++]; const float* boa = (const float*)d_in[pi++];
  const float* Wov = (const float*)d_in[pi++];
  const float* ga = (const float*)d_in[pi++]; const float* ba = (const float*)d_in[pi++];
  const float* gv = (const float*)d_in[pi++];

  float* ws = (float*)d_ws;
  size_t off = 0;
  auto take = [&](size_t n) { float* p = ws + off; off += n; return p; };
  float* EA   = take(BN_ * A_);
  float* EVb  = take(BN_ * Vd_ * 3);
  float* EMBA = take(BN_ * A_);
  float* EMBV = take(BN_ * Vd_ * 3);
  float* YA   = take(BN_ * A_);
  float* YV   = take(BN_ * Vd_ * 3);
  float* T1   = take(BN_ * A_);
  float* T2   = take(BN_ * A_);
  float* TV   = take(BN_ * Vd_ * 3);
  float* LVb  = take(BN_ * Vd_ * 3);
  float* ARb  = take(BN_ * RANK_);
  float* V1b  = take(BN_ * RANK_ * 3);
  float* V2b  = take(BN_ * RANK_ * 3);
  float* PAIN = take(BN_ * RANK_);
  float* PAB  = take(BN_ * A_);
  float* PVIN = take(BN_ * RANK_ * 3);
  float* PVB  = take(BN_ * Vd_ * 3);
  float* PQb  = take(BN_ * 6);
  float* PKb  = take(BN_ * 6);
  float* SQQ  = take(BN_);
  float* SQK  = take(BN_);
  float* QB   = take(BN_ * 256);
  float* KB   = take(BN_ * 256);
  float* VAB  = take(BN_ * A_);
  float* VVB  = take(BN_ * Vd_ * 3);
  float* OAP  = take(BN_ * A_);
  float* OVP  = take(BN_ * Vd_ * 3);
  float* CA   = take(BN_ * A_);
  float* CV   = take(BN_ * Vd_ * 3);

  float* out_p0 = (float*)d_out;
  float* out_p1 = out_p0 + BN_ * 3;
  float* out_xa = out_p1 + BN_ * 3;
  float* out_xv = out_xa + BN_ * A_;

  const int GR = (int)(BN_ / 16);  // 512 row-tiles

  hipMemcpyAsync(out_p0, pos0, BN_ * 3 * sizeof(float), hipMemcpyDeviceToDevice, stream);
  hipMemcpyAsync(out_p1, pos1, BN_ * 3 * sizeof(float), hipMemcpyDeviceToDevice, stream);

  edge_feat_kernel<<<B_ * (N_ - 1), 128, 0, stream>>>(pos0, pos1, We1, be1, We2, be2, Wev, EA, EVb);
  emb_kernel<<<BN_, 128, 0, stream>>>(pos0, pos1, EA, EVb, Wn1, bn1, Wn2, bn2, Wnv, EMBA, EMBV);
  conv_a_kernel<<<dim3(GR, 2), 128, 0, stream>>>(x_a, K0a, b0a, EMBA, YA);
  conv_v_kernel<<<GR, 128, 0, stream>>>(x_v, K0v, EMBV, YV);

  gemm_kernel<<<GR, 128, 0, stream>>>(YA, A_, Wl1, bl1, T1, A_, A_, A_, 1, 0);
  gemm_kernel<<<GR, 128, 0, stream>>>(T1, A_, Wl2, bl2, T2, A_, A_, A_, 1, 0);
  gemm_kernel<<<GR, 128, 0, stream>>>(T2, A_, Wl3, bl3, T1, A_, A_, A_, 0, 0);  // T1 = la
  vlin_kernel<<<BN_, 64, 0, stream>>>(YV, Wv1, TV, Vd_, Vd_, 0, 0);
  vlin_kernel<<<BN_, 64, 0, stream>>>(TV, Wv2, LVb, Vd_, Vd_, 1, 0);
  vlin_kernel<<<BN_, 64, 0, stream>>>(LVb, Wv3, TV, Vd_, Vd_, 1, 0);            // TV = lv
  gemm_kernel<<<GR, 128, 0, stream>>>(YA, A_, War, bar, ARb, RANK_, A_, RANK_, 0, 0);
  vlin_kernel<<<BN_, 64, 0, stream>>>(YV, Wvr1, V1b, Vd_, RANK_, 0, 0);
  vlin_kernel<<<BN_, 64, 0, stream>>>(YV, Wvr2, V2b, Vd_, RANK_, 0, 0);
  rank_mix_kernel<<<BN_, 32, 0, stream>>>(ARb, V1b, V2b, PAIN, PVIN);
  gemm_kernel<<<GR, 128, 0, stream>>>(PAIN, RANK_, Wpa, bpa, PAB, A_, RANK_, A_, 0, 0);
  vlin_kernel<<<BN_, 64, 0, stream>>>(PVIN, Wpv, PVB, RANK_, Vd_, 0, 0);
  sgn_kernel<<<BN_, 128, 0, stream>>>(T1, PAB, glra, blra, nullptr, YA);        // y_a += sgn(la+pa)
  vgn_kernel<<<BN_, 64, 0, stream>>>(TV, PVB, glrv, nullptr, YV);               // y_v += vgn(lv+pv)

  probes_kernel<<<BN_, 32, 0, stream>>>(YA, YV, pos0, pos1, Wprobe_a, Wprobe_v, PQb, PKb, SQQ, SQK);
  gemm_kernel<<<GR, 128, 0, stream>>>(YA, A_, Wq, nullptr, QB, 256, A_, 256, 0, 0);
  gemm_kernel<<<GR, 128, 0, stream>>>(YA, A_, Wk, nullptr, KB, 256, A_, 256, 0, 0);
  gemm_kernel<<<GR, 128, 0, stream>>>(YA, A_, Wva, nullptr, VAB, A_, A_, A_, 0, 0);
  vlin_kernel<<<BN_, 64, 0, stream>>>(YV, Wvv, VVB, Vd_, Vd_, 0, 0);
  attn_kernel<<<B_ * H_ * (N_ / 16), 128, 0, stream>>>(QB, KB, VAB, VVB, PQb, PKb, SQQ, SQK, OAP, OVP);
  gemm_kernel<<<GR, 128, 0, stream>>>(OAP, A_, Woa, boa, YA, A_, A_, A_, 0, 1); // y_a += oa
  vlin_kernel<<<BN_, 64, 0, stream>>>(OVP, Wov, YV, Vd_, Vd_, 0, 1);            // y_v += ov

  conv_a_kernel<<<dim3(GR, 2), 128, 0, stream>>>(YA, K1a, b1a, nullptr, CA);
  conv_v_kernel<<<GR, 128, 0, stream>>>(YV, K1v, nullptr, CV);
  sgn_kernel<<<BN_, 128, 0, stream>>>(CA, nullptr, ga, ba, x_a, out_xa);        // x_a + sgn(conv)
  vgn_kernel<<<BN_, 64, 0, stream>>>(CV, nullptr, gv, x_v, out_xv);             // x_v + vgn(conv)
}